// GraphClassifier_26749056320117
// MI455X (gfx1250) — compile-verified
//
#include <hip/hip_runtime.h>

#define N_NODES     50000
#define N_EDGES     800000
#define HIDDEN      128
#define NUM_CLASSES 10
#define NUM_GRAPHS  32

typedef float v2f __attribute__((ext_vector_type(2)));
typedef float v8f __attribute__((ext_vector_type(8)));

// ---------------------------------------------------------------------------
// Zero-fill (used for agg / pooling scratch each call; harness does not
// re-zero workspace between replays).
// ---------------------------------------------------------------------------
__global__ __launch_bounds__(256) void fill_zero(float* __restrict__ p, int n) {
    int i = blockIdx.x * blockDim.x + threadIdx.x;
    int stride = gridDim.x * blockDim.x;
    for (; i < n; i += stride) p[i] = 0.0f;
}

// ---------------------------------------------------------------------------
// out[n_rows x 128] = A1 @ W1^T (+ A2 @ W2^T) + bias, optional ReLU.
// One wave32 per 16x16 output tile, 8 waves per block (block = 16 rows x 128
// cols). Uses V_WMMA_F32_16X16X4_F32 (fp32 in / fp32 accum), K stepped by 4.
//
// Operand layouts per CDNA5 ISA 7.12.2:
//   A 16x4 f32 : lanes 0-15 -> K={0,1}, lanes 16-31 -> K={2,3}, M = lane&15
//   B  4x16 f32: rows (K) striped across VGPRs, lane&15 = N, half = K pair
//   C/D 16x16  : VGPR v -> M = v + 8*(lane>=16), N = lane&15
// ---------------------------------------------------------------------------
__global__ __launch_bounds__(256) void gemm_wmma_f32(
    const float* __restrict__ A1, const float* __restrict__ W1,
    const float* __restrict__ A2, const float* __restrict__ W2,
    const float* __restrict__ bias, float* __restrict__ out,
    int relu)
{
    const int tile = blockIdx.x * 8 + (threadIdx.x >> 5);
    const int lane = threadIdx.x & 31;
    const int half = lane >> 4;      // 0: K={0,1}  1: K={2,3}
    const int r    = lane & 15;
    const int m0   = (tile >> 3) * 16;   // 8 col-tiles across HIDDEN=128
    const int n0   = (tile & 7) * 16;

    v8f c = {0.f, 0.f, 0.f, 0.f, 0.f, 0.f, 0.f, 0.f};

    {
        const float* a = A1 + (size_t)(m0 + r) * HIDDEN + 2 * half;
        const float* b = W1 + (size_t)(n0 + r) * HIDDEN + 2 * half;
        #pragma unroll 8
        for (int k = 0; k < HIDDEN; k += 4) {
            v2f av = *(const v2f*)(a + k);
            v2f bv = *(const v2f*)(b + k);
            c = __builtin_amdgcn_wmma_f32_16x16x4_f32(
                    false, av, false, bv, (short)0, c, false, false);
        }
    }
    if (A2 != nullptr) {  // uniform (SGPR) branch: EXEC stays all-ones for WMMA
        const float* a = A2 + (size_t)(m0 + r) * HIDDEN + 2 * half;
        const float* b = W2 + (size_t)(n0 + r) * HIDDEN + 2 * half;
        #pragma unroll 8
        for (int k = 0; k < HIDDEN; k += 4) {
            v2f av = *(const v2f*)(a + k);
            v2f bv = *(const v2f*)(b + k);
            c = __builtin_amdgcn_wmma_f32_16x16x4_f32(
                    false, av, false, bv, (short)0, c, false, false);
        }
    }

    const int   col = n0 + r;
    const float bv  = bias ? bias[col] : 0.0f;
    #pragma unroll
    for (int v = 0; v < 8; ++v) {
        const int row = m0 + v + 8 * half;
        float val = c[v] + bv;
        if (relu) val = fmaxf(val, 0.0f);
        out[(size_t)row * HIDDEN + col] = val;
    }
}

// ---------------------------------------------------------------------------
// agg[dst[e]] += edge_weight[e] * h[src[e]]   (one wave per edge, float4/lane)
// ---------------------------------------------------------------------------
__global__ __launch_bounds__(256) void edge_message_scatter(
    const float* __restrict__ h, const int* __restrict__ src,
    const int* __restrict__ dst, const float* __restrict__ ew,
    float* __restrict__ agg, int n_edges)
{
    const int e    = (blockIdx.x * 256 + threadIdx.x) >> 5;
    const int lane = threadIdx.x & 31;
    if (e >= n_edges) return;                 // uniform per wave
    const int   s = src[e];
    const int   d = dst[e];
    const float w = ew[e];
    const float4 v = ((const float4*)(h + (size_t)s * HIDDEN))[lane];
    float* ap = agg + (size_t)d * HIDDEN + lane * 4;
    atomicAdd(ap + 0, v.x * w);
    atomicAdd(ap + 1, v.y * w);
    atomicAdd(ap + 2, v.z * w);
    atomicAdd(ap + 3, v.w * w);
}

// ---------------------------------------------------------------------------
// g_sum[batch[i]] += h[i]; cnt[batch[i]] += 1   (one wave per node)
// ---------------------------------------------------------------------------
__global__ __launch_bounds__(256) void pool_scatter(
    const float* __restrict__ h, const int* __restrict__ batch,
    float* __restrict__ g_sum, float* __restrict__ cnt, int n_nodes)
{
    const int node = (blockIdx.x * 256 + threadIdx.x) >> 5;
    const int lane = threadIdx.x & 31;
    if (node >= n_nodes) return;
    const int b = batch[node];
    const float4 v = ((const float4*)(h + (size_t)node * HIDDEN))[lane];
    float* gp = g_sum + (size_t)b * HIDDEN + lane * 4;
    atomicAdd(gp + 0, v.x);
    atomicAdd(gp + 1, v.y);
    atomicAdd(gp + 2, v.z);
    atomicAdd(gp + 3, v.w);
    if (lane == 0) atomicAdd(cnt + b, 1.0f);
}

// ---------------------------------------------------------------------------
// g = g_sum / max(cnt,1) -> out[320:]; logits = g @ cls_w^T + cls_b -> out[:320]
// One block per graph.
// ---------------------------------------------------------------------------
__global__ __launch_bounds__(128) void finalize_kernel(
    const float* __restrict__ g_sum, const float* __restrict__ cnt,
    const float* __restrict__ cls_w, const float* __restrict__ cls_b,
    float* __restrict__ out)
{
    __shared__ float gl[HIDDEN];
    const int gr = blockIdx.x;
    const int t  = threadIdx.x;
    const float c  = fmaxf(cnt[gr], 1.0f);
    const float gv = g_sum[(size_t)gr * HIDDEN + t] / c;
    out[NUM_GRAPHS * NUM_CLASSES + gr * HIDDEN + t] = gv;  // g output
    gl[t] = gv;
    __syncthreads();
    if (t < NUM_CLASSES) {
        float acc = cls_b[t];
        #pragma unroll 8
        for (int k = 0; k < HIDDEN; ++k) acc += gl[k] * cls_w[t * HIDDEN + k];
        out[gr * NUM_CLASSES + t] = acc;                   // logits output
    }
}

// ---------------------------------------------------------------------------
extern "C" void kernel_launch(void* const* d_in, const int* in_sizes, int n_in,
                              void* d_out, int out_size, void* d_ws, size_t ws_size,
                              hipStream_t stream)
{
    const float* x_nodes     = (const float*)d_in[0];
    const int*   edge_index  = (const int*)  d_in[1];
    const float* edge_weight = (const float*)d_in[2];
    const int*   batch       = (const int*)  d_in[3];
    const float* proj_w      = (const float*)d_in[4];
    const float* proj_b      = (const float*)d_in[5];
    const float* rel_w       = (const float*)d_in[6];
    const float* rel_b       = (const float*)d_in[7];
    const float* root_w      = (const float*)d_in[8];
    const float* cls_w       = (const float*)d_in[9];
    const float* cls_b       = (const float*)d_in[10];
    float* out = (float*)d_out;

    const int* src = edge_index;            // edge_index[0, :]
    const int* dst = edge_index + N_EDGES;  // edge_index[1, :]

    const size_t NH = (size_t)N_NODES * HIDDEN;   // 6.4M floats
    float* h0    = (float*)d_ws;
    float* agg   = h0  + NH;
    float* h1    = agg + NH;
    float* g_sum = h1  + NH;
    float* cnt   = g_sum + NUM_GRAPHS * HIDDEN;   // contiguous with g_sum

    const int gemm_blocks = (N_NODES / 16) * (HIDDEN / 16) / 8;  // 3125
    const int edge_blocks = (N_EDGES * 32 + 255) / 256;          // 100000
    const int node_blocks = (N_NODES * 32 + 255) / 256;          // 6250

    // h = x @ proj_w^T + proj_b
    gemm_wmma_f32<<<gemm_blocks, 256, 0, stream>>>(
        x_nodes, proj_w, nullptr, nullptr, proj_b, h0, 0);

    const float* hin = h0;
    float* hout = h1;
    for (int l = 0; l < 2; ++l) {
        fill_zero<<<1024, 256, 0, stream>>>(agg, (int)NH);
        edge_message_scatter<<<edge_blocks, 256, 0, stream>>>(
            hin, src, dst, edge_weight, agg, N_EDGES);
        // h = relu(agg @ rel_w[l]^T + rel_b[l] + h @ root_w[l]^T)
        gemm_wmma_f32<<<gemm_blocks, 256, 0, stream>>>(
            agg, rel_w + (size_t)l * HIDDEN * HIDDEN,
            hin, root_w + (size_t)l * HIDDEN * HIDDEN,
            rel_b + l * HIDDEN, hout, 1);
        const float* tmp = hin; hin = hout; hout = (float*)tmp;
    }

    // global mean pool + classifier
    fill_zero<<<32, 256, 0, stream>>>(g_sum, NUM_GRAPHS * HIDDEN + NUM_GRAPHS);
    pool_scatter<<<node_blocks, 256, 0, stream>>>(hin, batch, g_sum, cnt, N_NODES);
    finalize_kernel<<<NUM_GRAPHS, 128, 0, stream>>>(g_sum, cnt, cls_w, cls_b, out);
}